// LocalizationLayer_85246510891783
// MI455X (gfx1250) — compile-verified
//
#include <hip/hip_runtime.h>
#include <hip/hip_bf16.h>
#include <math.h>

typedef unsigned int u32;
typedef unsigned long long u64;
typedef u32 u32x4v __attribute__((ext_vector_type(4)));
typedef u32 u32x8v __attribute__((ext_vector_type(8)));

#define NBOX   8192
#define NWORDS 256          // 8192 / 32 bits
#define NMS_THRESH 0.7f
#define DEPTH  8            // async prefetch depth (rows) in merge kernel

// ---------------------------------------------------------------------------
// Workspace layout (bytes)
// ---------------------------------------------------------------------------
#define WS_XYXY      0          // float4[8192]  = 128K
#define WS_CXCYWH    131072     // float4[8192]  = 128K
#define WS_PROBS     262144     // float [8192]  =  32K
#define WS_ITEMS     294912     // u64   [8192]  =  64K
#define WS_XYXY_S    360448     // float4[8192]  = 128K
#define WS_CXCYWH_S  491520     // float4[8192]  = 128K
#define WS_PROBS_S   622592     // float [8192]  =  32K
#define WS_KEEPW     655360     // u32   [256]   =   1K
#define WS_MASK      1048576    // u32   [8192*256] = 8M

// ---------------------------------------------------------------------------
// K1: decode boxes, clip, sigmoid, pack sort keys
// ---------------------------------------------------------------------------
__global__ __launch_bounds__(256)
void prep_kernel(const float4* __restrict__ boxes, const float* __restrict__ scores,
                 const int* __restrict__ ih, const int* __restrict__ iw,
                 float4* __restrict__ xyxy, float4* __restrict__ cxcywh,
                 float* __restrict__ probs, u64* __restrict__ items) {
    int i = blockIdx.x * 256 + threadIdx.x;
    if (i >= NBOX) return;
    float4 b = boxes[i];
    float xm = (float)(iw[0] - 1);
    float ym = (float)(ih[0] - 1);
    float x1 = fminf(fmaxf(b.x - 0.5f * b.z, 0.0f), xm);
    float x2 = fminf(fmaxf(b.x + 0.5f * b.z, 0.0f), xm);
    float y1 = fminf(fmaxf(b.y - 0.5f * b.w, 0.0f), ym);
    float y2 = fminf(fmaxf(b.y + 0.5f * b.w, 0.0f), ym);
    float cw = x2 - x1, ch = y2 - y1;
    bool valid = (cw > 0.0f) && (ch > 0.0f);
    float p  = 1.0f / (1.0f + expf(-scores[i]));
    float mp = valid ? p : -INFINITY;
    xyxy[i]   = make_float4(x1, y1, x2, y2);
    cxcywh[i] = make_float4(0.5f * (x1 + x2), 0.5f * (y1 + y2), cw, ch);
    probs[i]  = mp;
    u32 u = __float_as_uint(mp);
    u32 s = (u & 0x80000000u) ? ~u : (u | 0x80000000u);   // ascending-order map
    items[i] = (((u64)(~s)) << 32) | (u32)i;              // ascending => desc prob
}

// ---------------------------------------------------------------------------
// K2: single-block bitonic sort of 8192 u64 items in 64KB LDS
// ---------------------------------------------------------------------------
__global__ __launch_bounds__(1024)
void sort_kernel(u64* __restrict__ items) {
    __shared__ u64 s[NBOX];
    int tid = threadIdx.x;
    for (int i = tid; i < NBOX; i += 1024) s[i] = items[i];
    __syncthreads();
    for (int k = 2; k <= NBOX; k <<= 1) {
        for (int j = k >> 1; j > 0; j >>= 1) {
            for (int i = tid; i < NBOX; i += 1024) {
                int ixj = i ^ j;
                if (ixj > i) {
                    u64 a = s[i], b = s[ixj];
                    bool up = ((i & k) == 0);
                    bool sw = up ? (a > b) : (a < b);
                    if (sw) { s[i] = b; s[ixj] = a; }
                }
            }
            __syncthreads();
        }
    }
    for (int i = tid; i < NBOX; i += 1024) items[i] = s[i];
}

// ---------------------------------------------------------------------------
// K3: gather boxes/probs into sorted order
// ---------------------------------------------------------------------------
__global__ __launch_bounds__(256)
void gather_kernel(const u64* __restrict__ items,
                   const float4* __restrict__ xyxy, const float4* __restrict__ cxcywh,
                   const float* __restrict__ probs,
                   float4* __restrict__ xyxy_s, float4* __restrict__ cxcywh_s,
                   float* __restrict__ probs_s) {
    int p = blockIdx.x * 256 + threadIdx.x;
    if (p >= NBOX) return;
    u32 o = (u32)(items[p] & 0xFFFFFFFFull);
    xyxy_s[p]   = xyxy[o];
    cxcywh_s[p] = cxcywh[o];
    probs_s[p]  = probs[o];
}

// ---------------------------------------------------------------------------
// K4: suppression mask. One wave32 per row; bit j set iff IoU(row,j)>0.7, j>row.
// Column-box tiles (4KB) staged via the Tensor Data Mover, double-buffered so
// the DMA for tile t+1 overlaps IoU compute on tile t (TENSORcnt-tracked).
// Tiles entirely below the block's rows are zero-filled without compute.
// ---------------------------------------------------------------------------
__global__ __launch_bounds__(256)
void mask_kernel(const float4* __restrict__ xyxy_s, u32* __restrict__ mask) {
    __shared__ float4 tiles[2][256];
    const int wave = threadIdx.x >> 5;
    const int lane = threadIdx.x & 31;
    const int row  = blockIdx.x * 8 + wave;

    float4 rb = xyxy_s[row];
    float rarea = (rb.z - rb.x) * (rb.w - rb.y);

    // tiles [0, tFirst) have all cols <= min row in block -> all-zero words
    const int tFirst = blockIdx.x >> 5;
    for (int w = lane; w < tFirst * 8; w += 32)
        mask[(size_t)row * NWORDS + w] = 0u;

    // TDM descriptor issue: 1D copy of 1024 f32 (one 256-box tile) to LDS
    auto issue = [&](int t, int buf) {
        u64 ga = (u64)(const void*)(xyxy_s + t * 256);
        u32 lds_off = (u32)(u64)(&tiles[buf][0]);
        u32x4v g0 = { 1u,                                  // count=1 valid descriptor
                      lds_off,                             // lds_addr
                      (u32)ga,                             // global_addr[31:0]
                      ((u32)(ga >> 32) & 0x01FFFFFFu) | 0x80000000u }; // ga[56:32] | type=2
        u32x8v g1 = { 0x00020000u,   // workgroup_mask=0, data_size=2 (4B)
                      0x04000000u,   // tensor_dim0[15:0]=1024 in bits[31:16]
                      0x00010000u,   // tensor_dim0[31:16]=0, tensor_dim1[15:0]=1
                      0x04000000u,   // tensor_dim1 hi=0, tile_dim0=1024 in bits[31:16]
                      0x00000001u,   // tile_dim1=1, tile_dim2=0
                      1024u,         // tensor_dim0_stride[31:0]
                      0u, 0u };
        asm volatile("tensor_load_to_lds %0, %1" :: "s"(g0), "s"(g1) : "memory");
    };

    if (wave == 0) { issue(tFirst, 0); __builtin_amdgcn_s_wait_tensorcnt(0); }
    __syncthreads();

    for (int t = tFirst; t < NBOX / 256; ++t) {
        const int buf = (t - tFirst) & 1;
        if (wave == 0 && (t + 1) < NBOX / 256) issue(t + 1, buf ^ 1); // overlap DMA
        const int colBase = t * 256;
        for (int c = 0; c < 8; ++c) {
            int col = colBase + c * 32 + lane;
            float4 cb = tiles[buf][c * 32 + lane];
            float carea = (cb.z - cb.x) * (cb.w - cb.y);
            float ix1 = fmaxf(rb.x, cb.x);
            float iy1 = fmaxf(rb.y, cb.y);
            float ix2 = fminf(rb.z, cb.z);
            float iy2 = fminf(rb.w, cb.w);
            float inter = fmaxf(ix2 - ix1, 0.0f) * fmaxf(iy2 - iy1, 0.0f);
            float uni   = fmaxf(rarea + carea - inter, 1e-8f);
            bool pred = (inter > NMS_THRESH * uni) && (col > row);
            u32 w = (u32)__ballot(pred);
            if (lane == 0) mask[(size_t)row * NWORDS + ((u32)(colBase + c * 32) >> 5)] = w;
        }
        if (wave == 0) __builtin_amdgcn_s_wait_tensorcnt(0);
        __syncthreads();
    }
}

// ---------------------------------------------------------------------------
// K5: serial greedy merge — ONE wave32, wave-synchronous, state in LDS.
// Mask rows streamed through an 8-deep LDS ring via async global->LDS loads:
// row i+8 is issued while row i is consumed; s_wait_asynccnt <=14 guarantees
// (in-order completion) that row i has landed; s_wait_dscnt 0 before reuse.
// ---------------------------------------------------------------------------
__global__ __launch_bounds__(32)
void merge_kernel(const u32* __restrict__ mask, const float* __restrict__ probs_s,
                  u32* __restrict__ keepw) {
    __shared__ u32 sup_lds[NWORDS];
    __shared__ u32 keep_lds[NWORDS];
    __shared__ u32 validw[NWORDS];
    __shared__ u32 ring[DEPTH * NWORDS];
    const int lane = threadIdx.x;

    for (int w = lane; w < NWORDS; w += 32) {
        sup_lds[w] = 0u; keep_lds[w] = 0u;
        u32 v = 0u;
        for (int b = 0; b < 32; ++b)
            v |= (probs_s[w * 32 + b] > -1.0f) ? (1u << b) : 0u;
        validw[w] = v;
    }
    __syncthreads();

    const u32 ring_base = (u32)(u64)(&ring[0]);

    // prologue: issue rows 0..DEPTH-1 into the ring (2 x b128 per lane per row)
    for (int d = 0; d < DEPTH; ++d) {
        u32 l0 = ring_base + (u32)(d * NWORDS + lane * 8) * 4u;
        const u32* g0 = mask + (size_t)d * NWORDS + lane * 8;
        asm volatile("global_load_async_to_lds_b128 %0, %1, off" :: "v"(l0), "v"(g0) : "memory");
        asm volatile("global_load_async_to_lds_b128 %0, %1, off" :: "v"(l0 + 16u), "v"(g0 + 4) : "memory");
    }

    const uint4* ring4 = (const uint4*)ring;
    for (int i = 0; i < NBOX; ++i) {
        const int slot = i & (DEPTH - 1);
        // oldest outstanding pair (row i) must have landed: cnt <= 2*(DEPTH-1)
        asm volatile("s_wait_asynccnt 0xe" ::: "memory");
        uint4 a = ring4[slot * 64 + lane * 2];
        uint4 b = ring4[slot * 64 + lane * 2 + 1];
        const int w = i >> 5, bit = i & 31;
        bool kept = (((validw[w] >> bit) & 1u) != 0u) &&
                    (((sup_lds[w] >> bit) & 1u) == 0u);
        if (kept) {
            if (lane == 0) keep_lds[w] |= (1u << bit);
            int base = lane * 8;   // lane owns words [lane*8 .. lane*8+7]
            sup_lds[base + 0] |= a.x; sup_lds[base + 1] |= a.y;
            sup_lds[base + 2] |= a.z; sup_lds[base + 3] |= a.w;
            sup_lds[base + 4] |= b.x; sup_lds[base + 5] |= b.y;
            sup_lds[base + 6] |= b.z; sup_lds[base + 7] |= b.w;
        }
        if (i + DEPTH < NBOX) {
            // all DS reads of this slot must retire before the async overwrite
            asm volatile("s_wait_dscnt 0x0" ::: "memory");
            u32 l0 = ring_base + (u32)(slot * NWORDS + lane * 8) * 4u;
            const u32* g0 = mask + (size_t)(i + DEPTH) * NWORDS + lane * 8;
            asm volatile("global_load_async_to_lds_b128 %0, %1, off" :: "v"(l0), "v"(g0) : "memory");
            asm volatile("global_load_async_to_lds_b128 %0, %1, off" :: "v"(l0 + 16u), "v"(g0 + 4) : "memory");
        }
    }
    __syncthreads();
    for (int w = lane; w < NWORDS; w += 32) keepw[w] = keep_lds[w];
}

// ---------------------------------------------------------------------------
// K6: zero output, prefix-sum keep bits, scatter first num_proposals kept boxes
// ---------------------------------------------------------------------------
__global__ __launch_bounds__(1024)
void emit_kernel(const u32* __restrict__ keepw, const float4* __restrict__ cxcywh_s,
                 const int* __restrict__ nprop_p, float4* __restrict__ out,
                 int out_rows) {
    __shared__ int pfx[NWORDS + 1];
    const int t = threadIdx.x;
    int nprop = nprop_p[0];
    if (nprop > out_rows) nprop = out_rows;

    for (int i = t; i < out_rows; i += 1024) out[i] = make_float4(0.f, 0.f, 0.f, 0.f);
    if (t == 0) {
        int s = 0;
        for (int w = 0; w < NWORDS; ++w) { pfx[w] = s; s += __popc(keepw[w]); }
        pfx[NWORDS] = s;
    }
    __syncthreads();

    for (int i = t; i < NBOX; i += 1024) {
        u32 wv = keepw[i >> 5];
        if ((wv >> (i & 31)) & 1u) {
            int rank = pfx[i >> 5] + __popc(wv & ((1u << (i & 31)) - 1u));
            if (rank < nprop) out[rank] = cxcywh_s[i];
        }
    }
}

// ---------------------------------------------------------------------------
// Launch
// ---------------------------------------------------------------------------
extern "C" void kernel_launch(void* const* d_in, const int* in_sizes, int n_in,
                              void* d_out, int out_size, void* d_ws, size_t ws_size,
                              hipStream_t stream) {
    const float4* boxes  = (const float4*)d_in[0];   // (1,8192,4) f32
    const float*  scores = (const float*)d_in[1];    // (1,8192,1) f32
    const int*    ih     = (const int*)d_in[2];
    const int*    iw     = (const int*)d_in[3];
    const int*    nprop  = (const int*)d_in[4];

    char* ws = (char*)d_ws;
    float4* xyxy     = (float4*)(ws + WS_XYXY);
    float4* cxcywh   = (float4*)(ws + WS_CXCYWH);
    float*  probs    = (float*)(ws + WS_PROBS);
    u64*    items    = (u64*)(ws + WS_ITEMS);
    float4* xyxy_s   = (float4*)(ws + WS_XYXY_S);
    float4* cxcywh_s = (float4*)(ws + WS_CXCYWH_S);
    float*  probs_s  = (float*)(ws + WS_PROBS_S);
    u32*    keepw    = (u32*)(ws + WS_KEEPW);
    u32*    mask     = (u32*)(ws + WS_MASK);

    prep_kernel<<<NBOX / 256, 256, 0, stream>>>(boxes, scores, ih, iw,
                                                xyxy, cxcywh, probs, items);
    sort_kernel<<<1, 1024, 0, stream>>>(items);
    gather_kernel<<<NBOX / 256, 256, 0, stream>>>(items, xyxy, cxcywh, probs,
                                                  xyxy_s, cxcywh_s, probs_s);
    mask_kernel<<<NBOX / 8, 256, 0, stream>>>(xyxy_s, mask);
    merge_kernel<<<1, 32, 0, stream>>>(mask, probs_s, keepw);
    emit_kernel<<<1, 1024, 0, stream>>>(keepw, cxcywh_s, nprop,
                                        (float4*)d_out, out_size / 4);
}